// SampledSoftmax_2448131359089
// MI455X (gfx1250) — compile-verified
//
#include <hip/hip_runtime.h>
#include <hip/hip_bf16.h>
#include <math.h>

// Problem constants (from the reference)
#define VOCAB        1000000
#define EMBED        128
#define BATCH        4096
#define NUM_SAMPLED  8192

#define LOG2E 1.4426950408889634f
#define LN2   0.6931471805599453f

typedef __attribute__((ext_vector_type(2))) float v2f;
typedef __attribute__((ext_vector_type(8))) float v8f;

// log(NUM_SAMPLED * P(id)), P(k) = log1p(1/(k+1)) / log(VOCAB+1)
__device__ __forceinline__ float log_correction(int id) {
    float k = (float)id;
    float p = log1pf(1.0f / (k + 1.0f)) / logf((float)VOCAB + 1.0f);
    return logf((float)NUM_SAMPLED * p);
}

__device__ __forceinline__ float fast_exp2(float x) {
    return __builtin_amdgcn_exp2f(x);
}
__device__ __forceinline__ float fast_log2(float x) {
    return __builtin_amdgcn_logf(x);
}

// ---------------------------------------------------------------------------
// Kernel 1: per-sampled-column bias, pre-scaled to log2 domain:
//   cb2[j] = (b[sid[j]] - log(NUM_SAMPLED*P)) * log2(e)
// ---------------------------------------------------------------------------
__global__ void ss_colbias_kernel(const int* __restrict__ sid,
                                  const float* __restrict__ b,
                                  float* __restrict__ cb2) {
    int j = blockIdx.x * blockDim.x + threadIdx.x;
    if (j < NUM_SAMPLED) {
        int id = sid[j];
        cb2[j] = (b[id] - log_correction(id)) * LOG2E;
    }
}

// ---------------------------------------------------------------------------
// Kernel 2: true logits (log2 domain), one wave32 per batch row.
// ---------------------------------------------------------------------------
__global__ void ss_truelogit_kernel(const float* __restrict__ pred,
                                    const int* __restrict__ labels,
                                    const float* __restrict__ W,
                                    const float* __restrict__ b,
                                    float* __restrict__ tl2) {
    int wave = threadIdx.x >> 5;
    int lane = threadIdx.x & 31;
    int row  = blockIdx.x * 8 + wave;
    if (row >= BATCH) return;

    int l = labels[row];
    const float4* p4 = (const float4*)(pred + (size_t)row * EMBED);
    const float4* w4 = (const float4*)(W + (size_t)l * EMBED);
    float4 a = p4[lane];
    float4 w = w4[lane];
    float s = a.x * w.x + a.y * w.y + a.z * w.z + a.w * w.w;
    #pragma unroll
    for (int m = 16; m >= 1; m >>= 1)
        s += __shfl_xor(s, m, 32);
    if (lane == 0)
        tl2[row] = (s + b[l] - log_correction(l)) * LOG2E;
}

// ---------------------------------------------------------------------------
// Kernel 3: fused GEMM + online logsumexp (log2 domain).
//   grid  = BATCH/16 = 256 blocks (one 16-row M tile per block)
//   block = 256 threads = 8 waves; wave w handles N-tile-pairs w, w+8, ...
// Per iteration: one 16x32 N slab = two independent accumulation chains
// (2-way WMMA ILP), 32 shared A fragments (64 VGPRs, loaded once), and a
// single lane-local logsumexp update folding both columns (no cross-lane
// ops, no shuffles, base-2 hardware exp). Register budget ~130 VGPRs so
// nothing spills.
// ---------------------------------------------------------------------------
__global__ void __launch_bounds__(256)
ss_gemm_lse_kernel(const float* __restrict__ pred,
                   const int* __restrict__ sid,
                   const float* __restrict__ W,
                   const float* __restrict__ cb2,
                   const float* __restrict__ tl2,
                   float* __restrict__ loss) {
    __shared__ float2 smem[8][16];   // [wave][row] (m, s) in log2 domain

    const int wave  = threadIdx.x >> 5;
    const int lane  = threadIdx.x & 31;
    const int laneN = lane & 15;      // N column within subtile / A row
    const int half  = lane >> 4;      // 0: K+0..1 rows v ; 1: K+2..3 rows v+8
    const int m0    = blockIdx.x * 16;

    // ---- A fragments: 32 x (16x4) f32 fragments = 64 VGPRs, loaded once ----
    const float* aptr = pred + (size_t)(m0 + laneN) * EMBED + half * 2;
    v2f afrag[32];
    #pragma unroll
    for (int kf = 0; kf < 32; ++kf)
        afrag[kf] = *(const v2f*)(aptr + kf * 4);

    // Lane-local running (m, s): slot v covers row v + 8*half; both N
    // subtiles of an iteration feed the same row stats.
    float runM[8], runS[8];
    #pragma unroll
    for (int v = 0; v < 8; ++v) { runM[v] = -INFINITY; runS[v] = 0.0f; }

    const int NPAIRS = NUM_SAMPLED / 32;   // 256 pairs of 16-wide subtiles
    for (int p = wave; p < NPAIRS; p += 8) {
        const int n0 = p * 32;
        const int idA = sid[n0 + laneN];
        const int idB = sid[n0 + 16 + laneN];
        const float* bptrA = W + (size_t)idA * EMBED + half * 2;
        const float* bptrB = W + (size_t)idB * EMBED + half * 2;
        const float colbA = cb2[n0 + laneN];
        const float colbB = cb2[n0 + 16 + laneN];

        v8f accA = {}, accB = {};
        #pragma unroll
        for (int kf = 0; kf < 32; ++kf) {
            v2f bA = *(const v2f*)(bptrA + kf * 4);
            v2f bB = *(const v2f*)(bptrB + kf * 4);
            accA = __builtin_amdgcn_wmma_f32_16x16x4_f32(
                false, afrag[kf], false, bA, (short)0, accA, false, false);
            accB = __builtin_amdgcn_wmma_f32_16x16x4_f32(
                false, afrag[kf], false, bB, (short)0, accB, false, false);
        }

        // Lane-local online logsumexp: fold both columns in one update.
        #pragma unroll
        for (int v = 0; v < 8; ++v) {
            float x0 = fmaf(accA[v], LOG2E, colbA);
            float x1 = fmaf(accB[v], LOG2E, colbB);
            float nm = fmaxf(fmaxf(runM[v], x0), x1);   // -> v_max3_num_f32
            runS[v] = runS[v] * fast_exp2(runM[v] - nm)
                    + fast_exp2(x0 - nm) + fast_exp2(x1 - nm);
            runM[v] = nm;
        }
    }

    // ---- Merge the 16 lanes of each half ((m,s)-merge, 4 xor steps) ----
    #pragma unroll
    for (int v = 0; v < 8; ++v) {
        #pragma unroll
        for (int m = 8; m >= 1; m >>= 1) {
            float om = __shfl_xor(runM[v], m, 32);
            float os = __shfl_xor(runS[v], m, 32);
            float nm = fmaxf(runM[v], om);
            runS[v] = runS[v] * fast_exp2(runM[v] - nm)
                    + os      * fast_exp2(om      - nm);
            runM[v] = nm;
        }
    }

    // ---- Publish per-wave stats (rows v / v+8 per half) ----
    if (laneN == 0) {
        #pragma unroll
        for (int v = 0; v < 8; ++v)
            smem[wave][v + 8 * half] = make_float2(runM[v], runS[v]);
    }
    __syncthreads();

    // ---- Merge 8 waves + true logit, deterministic fixed order ----
    if (threadIdx.x < 16) {
        int r = threadIdx.x;
        float M = -INFINITY, S = 0.0f;
        #pragma unroll
        for (int w = 0; w < 8; ++w) {
            float2 ms = smem[w][r];
            float Mn = fmaxf(M, ms.x);
            S = S * fast_exp2(M - Mn) + ms.y * fast_exp2(ms.x - Mn);
            M = Mn;
        }
        float t2 = tl2[m0 + r];
        float Mn = fmaxf(M, t2);
        S = S * fast_exp2(M - Mn) + fast_exp2(t2 - Mn);
        // loss = ln2 * (log2-domain lse - log2-domain true logit)
        loss[m0 + r] = LN2 * (Mn + fast_log2(S) - t2);
    }
}

// ---------------------------------------------------------------------------
// Kernel 4: deterministic mean over the 4096 per-row losses.
// ---------------------------------------------------------------------------
__global__ void ss_reduce_kernel(const float* __restrict__ loss,
                                 float* __restrict__ out) {
    __shared__ float sm[256];
    float s = 0.0f;
    for (int i = threadIdx.x; i < BATCH; i += 256)
        s += loss[i];
    sm[threadIdx.x] = s;
    __syncthreads();
    for (int off = 128; off > 0; off >>= 1) {
        if (threadIdx.x < off) sm[threadIdx.x] += sm[threadIdx.x + off];
        __syncthreads();
    }
    if (threadIdx.x == 0) out[0] = sm[0] * (1.0f / (float)BATCH);
}

// ---------------------------------------------------------------------------
extern "C" void kernel_launch(void* const* d_in, const int* in_sizes, int n_in,
                              void* d_out, int out_size, void* d_ws, size_t ws_size,
                              hipStream_t stream) {
    const float* pred   = (const float*)d_in[0];   // (4096,128) f32
    const int*   labels = (const int*)d_in[1];     // (4096,)   int
    const int*   sid    = (const int*)d_in[2];     // (8192,)   int
    const float* W      = (const float*)d_in[3];   // (1000000,128) f32
    const float* b      = (const float*)d_in[4];   // (1000000,) f32
    float*       out    = (float*)d_out;

    // Workspace layout (floats): [cb2: 8192][tl2: 4096][loss: 4096]
    float* cb2  = (float*)d_ws;
    float* tl2  = cb2 + NUM_SAMPLED;
    float* loss = tl2 + BATCH;

    ss_colbias_kernel<<<(NUM_SAMPLED + 255) / 256, 256, 0, stream>>>(sid, b, cb2);
    ss_truelogit_kernel<<<BATCH / 8, 256, 0, stream>>>(pred, labels, W, b, tl2);
    ss_gemm_lse_kernel<<<BATCH / 16, 256, 0, stream>>>(pred, sid, W, cb2, tl2, loss);
    ss_reduce_kernel<<<1, 256, 0, stream>>>(loss, out);
}